// HANLayer_67877663146319
// MI455X (gfx1250) — compile-verified
//
#include <hip/hip_runtime.h>
#include <math.h>

// ---------------- problem constants (from reference) ----------------
#define IN_C 256
#define HEADS 4
#define CPH 64            // channels per head
#define OUTF (HEADS*CPH)  // 256 = H*C, also == IN_C here
#define NEG_SLOPE 0.2f

typedef __attribute__((ext_vector_type(2))) float v2f;
typedef __attribute__((ext_vector_type(8))) float v8f;

// =====================================================================
// Kernel 1: h = x @ W  via V_WMMA_F32_16X16X4_F32.
// One wave per 16x16 output tile; K loop 256 in steps of 4.
// A (16x4 f32): lanes 0-15 -> M=lane, VGPR0=K0,VGPR1=K1; lanes 16-31 -> K2,K3.
// B (4x16 f32): lanes 0-15 -> N=lane, VGPR0=K0,VGPR1=K1; lanes 16-31 -> K2,K3.
// C/D (16x16 f32): VGPR r: lanes 0-15 M=r, lanes 16-31 M=8+r; N=lane&15.
// =====================================================================
__global__ void gemm_wmma_kernel(const float* __restrict__ x,
                                 const float* __restrict__ W,
                                 float* __restrict__ h,
                                 int nrows) {
    const int wave = (int)((blockIdx.x * blockDim.x + threadIdx.x) >> 5);
    const int lane = threadIdx.x & 31;
    const int colTiles = OUTF / 16;           // 16
    const int tileM = wave / colTiles;
    const int tileN = wave % colTiles;
    if (tileM * 16 >= nrows) return;          // uniform per wave

    const int m0 = tileM * 16;
    const int n0 = tileN * 16;
    const int halfsel = lane >> 4;            // 0: K pair {0,1}, 1: {2,3}
    const int l15 = lane & 15;

    const float* __restrict__ arow = x + (size_t)(m0 + l15) * IN_C;

    v8f acc = {};
#pragma unroll 8
    for (int k0 = 0; k0 < IN_C; k0 += 4) {
        const int kb = k0 + 2 * halfsel;
        v2f a, b;
        a.x = arow[kb];
        a.y = arow[kb + 1];
        b.x = W[(size_t)kb * OUTF + n0 + l15];
        b.y = W[(size_t)(kb + 1) * OUTF + n0 + l15];
        acc = __builtin_amdgcn_wmma_f32_16x16x4_f32(
            /*neg_a=*/false, a, /*neg_b=*/false, b,
            /*c_mod=*/(short)0, acc, /*reuse_a=*/false, /*reuse_b=*/false);
    }

    const int mbase = m0 + 8 * halfsel;
#pragma unroll
    for (int r = 0; r < 8; ++r) {
        h[(size_t)(mbase + r) * OUTF + n0 + l15] = acc[r];
    }
}

// =====================================================================
// Kernel 2: a_src[n,h] = dot(h[n,h,:], att_src[h,:]); same for a_dst.
// One wave (32 lanes) per (node, head); each lane handles 2 channels.
// =====================================================================
__global__ void attn_kernel(const float* __restrict__ h,
                            const float* __restrict__ att_src,
                            const float* __restrict__ att_dst,
                            float* __restrict__ a_src,
                            float* __restrict__ a_dst,
                            int n_nodes) {
    const int node = blockIdx.x;
    if (node >= n_nodes) return;
    const int head = threadIdx.x >> 5;        // blockDim.x == 128 -> 4 waves
    const int lane = threadIdx.x & 31;

    const float* hv = h + (size_t)node * OUTF + head * CPH;
    const float* as = att_src + head * CPH;
    const float* ad = att_dst + head * CPH;

    float h0 = hv[lane], h1 = hv[lane + 32];
    float ssum = h0 * as[lane] + h1 * as[lane + 32];
    float dsum = h0 * ad[lane] + h1 * ad[lane + 32];
#pragma unroll
    for (int off = 16; off > 0; off >>= 1) {
        ssum += __shfl_xor(ssum, off, 32);
        dsum += __shfl_xor(dsum, off, 32);
    }
    if (lane == 0) {
        a_src[node * HEADS + head] = ssum;
        a_dst[node * HEADS + head] = dsum;
    }
}

// =====================================================================
// Kernel 3: init accumulators (harness poisons buffers).
// =====================================================================
__global__ void init_kernel(float* __restrict__ out,
                            float* __restrict__ m,
                            float* __restrict__ denom,
                            int n_nodes) {
    const int i = blockIdx.x * blockDim.x + threadIdx.x;
    if (i < n_nodes * CPH) out[i] = 0.0f;
    if (i < n_nodes * HEADS) { m[i] = -INFINITY; denom[i] = 0.0f; }
}

__device__ __forceinline__ float leaky(float v) {
    return v > 0.0f ? v : NEG_SLOPE * v;
}

// Float atomic max (any sign; int order == float order for >=0, reversed
// uint order for <0). Lowers to native global atomics.
__device__ __forceinline__ void atomicMaxF(float* addr, float val) {
    if (val >= 0.0f)
        atomicMax((int*)addr, __float_as_int(val));
    else
        atomicMin((unsigned int*)addr, __float_as_uint(val));
}

__device__ __forceinline__ void edge_endpoints(const int* __restrict__ ei,
                                               int e, int n_edges,
                                               int& src, int& dst) {
    if (e < n_edges) { src = ei[e]; dst = ei[n_edges + e]; }
    else             { src = dst = e - n_edges; }   // self loops
}

// =====================================================================
// Kernel 4: per-(edge,head) leaky-relu logit -> segment max over dst.
// =====================================================================
__global__ void edge_max_kernel(const int* __restrict__ ei,
                                const float* __restrict__ a_src,
                                const float* __restrict__ a_dst,
                                float* __restrict__ m,
                                int n_edges, int e_tot) {
    const int t = blockIdx.x * blockDim.x + threadIdx.x;
    const int e = t >> 2, head = t & 3;
    if (e >= e_tot) return;
    int src, dst; edge_endpoints(ei, e, n_edges, src, dst);
    const float lg = leaky(a_src[src * HEADS + head] + a_dst[dst * HEADS + head]);
    atomicMaxF(&m[dst * HEADS + head], lg);
}

// =====================================================================
// Kernel 5: ex = exp(e - m[dst,h]); denom[dst,h] += ex.
// If alpha != nullptr, also stash un-normalized ex per (edge,head).
// =====================================================================
__global__ void edge_sum_kernel(const int* __restrict__ ei,
                                const float* __restrict__ a_src,
                                const float* __restrict__ a_dst,
                                const float* __restrict__ m,
                                float* __restrict__ denom,
                                float* __restrict__ alpha,   // may be null
                                int n_edges, int e_tot) {
    const int t = blockIdx.x * blockDim.x + threadIdx.x;
    const int e = t >> 2, head = t & 3;
    if (e >= e_tot) return;
    int src, dst; edge_endpoints(ei, e, n_edges, src, dst);
    const float lg = leaky(a_src[src * HEADS + head] + a_dst[dst * HEADS + head]);
    const float ex = __expf(lg - m[dst * HEADS + head]);
    if (alpha) alpha[t] = ex;
    atomicAdd(&denom[dst * HEADS + head], ex);
}

// =====================================================================
// Kernel 5b: normalize alpha in place: alpha /= (denom[dst,h] + eps).
// =====================================================================
__global__ void alpha_norm_kernel(const int* __restrict__ ei,
                                  const float* __restrict__ denom,
                                  float* __restrict__ alpha,
                                  int n_edges, int e_tot) {
    const int t = blockIdx.x * blockDim.x + threadIdx.x;
    const int e = t >> 2, head = t & 3;
    if (e >= e_tot) return;
    int src, dst; edge_endpoints(ei, e, n_edges, src, dst);
    (void)src;
    alpha[t] = alpha[t] / (denom[dst * HEADS + head] + 1e-16f);
}

// =====================================================================
// Kernel 6a (fast path): out[dst,c] += (1/H) * sum_h alpha[e,h]*h[src,h,c]
// Thread per (edge, channel); pure data movement (no transcendentals):
// 4 broadcast alpha loads + 4 coalesced 64-float gathers + 1 atomic.
// =====================================================================
__global__ void edge_scatter_alpha_kernel(const int* __restrict__ ei,
                                          const float* __restrict__ alpha,
                                          const float* __restrict__ h,
                                          float* __restrict__ out,
                                          int n_edges, int e_tot) {
    const int t = blockIdx.x * blockDim.x + threadIdx.x;
    const int e = t >> 6, c = t & 63;
    if (e >= e_tot) return;
    int src, dst; edge_endpoints(ei, e, n_edges, src, dst);

    const float* __restrict__ hp = h + (size_t)src * OUTF + c;
    const float* __restrict__ ap = alpha + (size_t)e * HEADS;
    float sum = 0.0f;
#pragma unroll
    for (int hh = 0; hh < HEADS; ++hh)
        sum += ap[hh] * hp[hh * CPH];
    atomicAdd(&out[(size_t)dst * CPH + c], sum * (1.0f / HEADS));
}

// =====================================================================
// Kernel 6b (fallback, no alpha workspace): recompute alpha inline.
// =====================================================================
__global__ void edge_scatter_kernel(const int* __restrict__ ei,
                                    const float* __restrict__ a_src,
                                    const float* __restrict__ a_dst,
                                    const float* __restrict__ m,
                                    const float* __restrict__ denom,
                                    const float* __restrict__ h,
                                    float* __restrict__ out,
                                    int n_edges, int e_tot) {
    const int t = blockIdx.x * blockDim.x + threadIdx.x;
    const int e = t >> 6, c = t & 63;
    if (e >= e_tot) return;
    int src, dst; edge_endpoints(ei, e, n_edges, src, dst);

    float sum = 0.0f;
#pragma unroll
    for (int hh = 0; hh < HEADS; ++hh) {
        const float lg = leaky(a_src[src * HEADS + hh] + a_dst[dst * HEADS + hh]);
        const float al = __expf(lg - m[dst * HEADS + hh]) /
                         (denom[dst * HEADS + hh] + 1e-16f);
        sum += al * h[(size_t)src * OUTF + hh * CPH + c];
    }
    atomicAdd(&out[(size_t)dst * CPH + c], sum * (1.0f / HEADS));
}

// =====================================================================
// Kernel 7: out += bias (broadcast over nodes)
// =====================================================================
__global__ void bias_kernel(float* __restrict__ out,
                            const float* __restrict__ bias,
                            int n_nodes) {
    const int i = blockIdx.x * blockDim.x + threadIdx.x;
    if (i < n_nodes * CPH) out[i] += bias[i & (CPH - 1)];
}

// =====================================================================
extern "C" void kernel_launch(void* const* d_in, const int* in_sizes, int n_in,
                              void* d_out, int out_size, void* d_ws, size_t ws_size,
                              hipStream_t stream) {
    const float* x       = (const float*)d_in[0];
    const int*   ei      = (const int*)d_in[1];
    const float* W       = (const float*)d_in[2];
    const float* att_src = (const float*)d_in[3];
    const float* att_dst = (const float*)d_in[4];
    const float* bias    = (const float*)d_in[5];
    float* out = (float*)d_out;

    const int n_nodes = in_sizes[0] / IN_C;     // 100000
    const int n_edges = in_sizes[1] / 2;        // 1600000
    const int e_tot   = n_edges + n_nodes;      // + self loops

    // workspace layout (floats)
    float* h      = (float*)d_ws;                      // N * 256
    float* a_src  = h      + (size_t)n_nodes * OUTF;   // N * H
    float* a_dst  = a_src  + (size_t)n_nodes * HEADS;
    float* m      = a_dst  + (size_t)n_nodes * HEADS;
    float* denom  = m      + (size_t)n_nodes * HEADS;
    float* alpha  = denom  + (size_t)n_nodes * HEADS;  // e_tot * H (optional)

    const size_t base_floats  = (size_t)n_nodes * OUTF + 4u * (size_t)n_nodes * HEADS;
    const size_t alpha_floats = (size_t)e_tot * HEADS;
    // deterministic per-run: ws_size is fixed by the harness
    const bool use_alpha = ws_size >= (base_floats + alpha_floats) * sizeof(float);
    (void)n_in; (void)out_size;

    // 1) GEMM h = x @ W (one wave per 16x16 tile; 8 waves per block)
    {
        const int tiles = (n_nodes / 16) * (OUTF / 16);   // N % 16 == 0
        const int blocks = (tiles + 7) / 8;               // 8 waves / 256 thr
        gemm_wmma_kernel<<<blocks, 256, 0, stream>>>(x, W, h, n_nodes);
    }

    // 2) attention logits per (node, head)
    attn_kernel<<<n_nodes, 128, 0, stream>>>(h, att_src, att_dst, a_src, a_dst, n_nodes);

    // 3) init out / m / denom
    {
        const int n = n_nodes * CPH;
        init_kernel<<<(n + 255) / 256, 256, 0, stream>>>(out, m, denom, n_nodes);
    }

    // 4) segment max
    {
        const int n = e_tot * HEADS;
        edge_max_kernel<<<(n + 255) / 256, 256, 0, stream>>>(ei, a_src, a_dst, m,
                                                             n_edges, e_tot);
    }

    // 5) segment sum of exp (+ stash un-normalized alpha if space)
    {
        const int n = e_tot * HEADS;
        edge_sum_kernel<<<(n + 255) / 256, 256, 0, stream>>>(
            ei, a_src, a_dst, m, denom, use_alpha ? alpha : nullptr,
            n_edges, e_tot);
    }

    // 5b/6) normalize + scatter (fast path), or fused recompute (fallback)
    if (use_alpha) {
        const int n = e_tot * HEADS;
        alpha_norm_kernel<<<(n + 255) / 256, 256, 0, stream>>>(ei, denom, alpha,
                                                               n_edges, e_tot);
        const long long ns = (long long)e_tot * CPH;
        edge_scatter_alpha_kernel<<<(int)((ns + 255) / 256), 256, 0, stream>>>(
            ei, alpha, h, out, n_edges, e_tot);
    } else {
        const long long ns = (long long)e_tot * CPH;
        edge_scatter_kernel<<<(int)((ns + 255) / 256), 256, 0, stream>>>(
            ei, a_src, a_dst, m, denom, h, out, n_edges, e_tot);
    }

    // 7) bias
    {
        const int n = n_nodes * CPH;
        bias_kernel<<<(n + 255) / 256, 256, 0, stream>>>(out, bias, n_nodes);
    }
}